// RGCN_56066503082348
// MI455X (gfx1250) — compile-verified
//
#include <hip/hip_runtime.h>
#include <hip/hip_bf16.h>

// ---------------- problem constants (match reference) ----------------
#define NN   100000
#define EE   1600000
#define RR   8
#define BB   8
#define FF   128          // IN == HID == OUT == 128
#define KA   (RR * FF)    // 1024, stacked relation-K
#define BN_EPS 1e-5f

typedef float v2f __attribute__((ext_vector_type(2)));
typedef float v8f __attribute__((ext_vector_type(8)));

// ---------------- utility: grid-stride zero ----------------
__global__ void k_zero(float* __restrict__ p, size_t n) {
    size_t i = (size_t)blockIdx.x * blockDim.x + threadIdx.x;
    size_t stride = (size_t)gridDim.x * blockDim.x;
    for (; i < n; i += stride) p[i] = 0.0f;
}

// ---------------- W_stacked[(r*128+i)*128+o] = sum_b comp[r,b]*basis[b,i,o] ----------------
__global__ void k_build_w(const float* __restrict__ basis,   // [B,128,128]
                          const float* __restrict__ comp,    // [R,B]
                          float* __restrict__ wst) {         // [1024,128]
    int idx = blockIdx.x * blockDim.x + threadIdx.x;         // 0 .. R*128*128-1
    if (idx >= RR * FF * FF) return;
    int r   = idx / (FF * FF);
    int rem = idx - r * FF * FF;          // i*128 + o
    float acc = 0.0f;
#pragma unroll
    for (int b = 0; b < BB; ++b)
        acc += comp[r * BB + b] * basis[(size_t)b * FF * FF + rem];
    wst[idx] = acc;
}

// ---------------- per-(dst,rel) edge counts ----------------
__global__ void k_count(const int* __restrict__ dstv, const int* __restrict__ et,
                        float* __restrict__ cnt) {
    int e = blockIdx.x * blockDim.x + threadIdx.x;
    if (e >= EE) return;
    atomicAdd(&cnt[(size_t)dstv[e] * RR + et[e]], 1.0f);
}

// ---------------- segment-sum scatter: 32 threads/edge, float4 gather ----------------
__global__ void k_scatter(const float* __restrict__ x,
                          const int* __restrict__ srcv, const int* __restrict__ dstv,
                          const int* __restrict__ et, float* __restrict__ mean) {
    unsigned gid = blockIdx.x * blockDim.x + threadIdx.x;    // E*32 = 51.2M
    unsigned e = gid >> 5;
    unsigned q = gid & 31u;                                  // float4 index 0..31
    if (e >= EE) return;
    int s = srcv[e], d = dstv[e], r = et[e];
    const float4 v = *reinterpret_cast<const float4*>(x + (size_t)s * FF + q * 4);
    float* base = mean + ((size_t)d * RR + r) * FF + q * 4;
    atomicAdd(base + 0, v.x);
    atomicAdd(base + 1, v.y);
    atomicAdd(base + 2, v.z);
    atomicAdd(base + 3, v.w);
}

// ---------------- WMMA fp32 GEMM (double-buffered LDS, fused mean-divide) ----------------
// out[N,128] = (sum[N,1024] / cnt) @ Wstack[1024,128] + X[N,128] @ root[128,128] + bias
// block = 256 threads = 8 waves; block owns 16 rows x 128 cols; wave w -> 16x16 tile at col w*16.
// Unified K loop over 9 chunks of 128: chunks 0..7 = mean-sums (scaled by 1/max(cnt,1)) vs Wstack,
// chunk 8 = X vs root. Double buffering overlaps chunk c+1 staging with chunk c WMMAs.
// V_WMMA_F32_16X16X4_F32 layouts: A lanes 0-15 row m=lane K0/K1, lanes 16-31 row m=lane-16 K2/K3;
// B v0 = rows K0|K2, v1 = rows K1|K3, N = lane%16; C/D VGPR j -> rows j / j+8, N = lane%16.
__global__ __launch_bounds__(256) void k_gemm(const float* __restrict__ A,     // [N,1024] seg-sums
                                              const float* __restrict__ X,     // [N,128]
                                              const float* __restrict__ cnt,   // [N*R]
                                              const float* __restrict__ W,     // [1024,128]
                                              const float* __restrict__ Rt,    // [128,128]
                                              const float* __restrict__ bias,  // [128]
                                              float* __restrict__ out) {       // [N,128]
    __shared__ float As[2][16][132];                    // pad 132: stride%64==4 -> conflict-free
    const int row0 = blockIdx.x * 16;
    const int tid  = threadIdx.x;
    const int wave = tid >> 5;
    const int lane = tid & 31;
    const int col0 = wave * 16;
    const int hl   = lane >> 4;                         // 0 or 1
    const int ln   = lane & 15;

    auto load_chunk = [&](int c, int buf) {
        for (int i = tid; i < 512; i += 256) {          // 16 rows x 32 float4
            int r  = i >> 5;
            int cv = i & 31;
            float4 v;
            float scale;
            if (c < 8) {
                v = *reinterpret_cast<const float4*>(
                    A + (size_t)(row0 + r) * KA + c * FF + cv * 4);
                scale = 1.0f / fmaxf(cnt[(size_t)(row0 + r) * RR + c], 1.0f);
            } else {
                v = *reinterpret_cast<const float4*>(
                    X + (size_t)(row0 + r) * FF + cv * 4);
                scale = 1.0f;
            }
            float4 w = make_float4(v.x * scale, v.y * scale, v.z * scale, v.w * scale);
            *reinterpret_cast<float4*>(&As[buf][r][cv * 4]) = w;
        }
    };

    v8f acc = {};

    load_chunk(0, 0);
    __syncthreads();
    for (int c = 0; c < 9; ++c) {
        if (c < 8) load_chunk(c + 1, (c + 1) & 1);      // stage next chunk into other buffer
        const float* __restrict__ Bp = (c < 8) ? (W + (size_t)c * FF * FF) : Rt;
        const float (*Ab)[132] = As[c & 1];
#pragma unroll 8
        for (int k = 0; k < 128; k += 4) {
            v2f a, b;
            a.x = Ab[ln][k + hl * 2 + 0];
            a.y = Ab[ln][k + hl * 2 + 1];
            const int kg = k + hl * 2;
            b.x = Bp[(size_t)kg * FF + col0 + ln];
            b.y = Bp[(size_t)(kg + 1) * FF + col0 + ln];
            acc = __builtin_amdgcn_wmma_f32_16x16x4_f32(false, a, false, b,
                                                        (short)0, acc, false, false);
        }
        __syncthreads();                                // next-chunk stores visible; safe to reuse
    }

    // ---- store D + bias ----
    const float bv = bias[col0 + ln];
#pragma unroll
    for (int j = 0; j < 8; ++j) {
        int m = j + hl * 8;
        out[(size_t)(row0 + m) * FF + col0 + ln] = acc[j] + bv;
    }
}

// ---------------- batchnorm stats: stats[c]=sum, stats[128+c]=sumsq ----------------
__global__ __launch_bounds__(128) void k_bn_stats(const float* __restrict__ h,
                                                  float* __restrict__ stats) {
    int c = threadIdx.x;                  // 0..127
    float s = 0.0f, q = 0.0f;
    for (int n = blockIdx.x; n < NN; n += gridDim.x) {
        float v = h[(size_t)n * FF + c];
        s += v;
        q += v * v;
    }
    atomicAdd(&stats[c], s);
    atomicAdd(&stats[FF + c], q);
}

// ---------------- batchnorm apply + LeakyReLU(0.1), in place ----------------
__global__ void k_bn_apply(float* __restrict__ h, const float* __restrict__ stats,
                           const float* __restrict__ gamma, const float* __restrict__ beta) {
    size_t idx = (size_t)blockIdx.x * blockDim.x + threadIdx.x;
    if (idx >= (size_t)NN * FF) return;
    int c = (int)(idx & 127u);
    const float inv_n = 1.0f / (float)NN;
    float m   = stats[c] * inv_n;
    float var = stats[FF + c] * inv_n - m * m;
    float v = (h[idx] - m) * rsqrtf(var + BN_EPS) * gamma[c] + beta[c];
    h[idx] = v > 0.0f ? v : 0.1f * v;
}

// ---------------- row-wise L2 normalize, in place ----------------
__global__ __launch_bounds__(128) void k_l2norm(float* __restrict__ out) {
    __shared__ float red[128];
    int row = blockIdx.x;
    int c   = threadIdx.x;
    float v = out[(size_t)row * FF + c];
    red[c] = v * v;
    __syncthreads();
    for (int s = 64; s > 0; s >>= 1) {
        if (c < s) red[c] += red[c + s];
        __syncthreads();
    }
    float norm = fmaxf(sqrtf(red[0]), 1e-12f);
    out[(size_t)row * FF + c] = v / norm;
}

// ---------------- host-side layer driver ----------------
static inline void run_layer(const float* x_in, const float* basis, const float* comp,
                             const float* root, const float* bias, float* h_out,
                             const int* srcv, const int* dstv, const int* et,
                             float* mean, float* cnt, float* wst, hipStream_t stream) {
    const size_t mean_cnt_elems = (size_t)NN * RR * FF + (size_t)NN * RR;
    k_zero<<<8192, 256, 0, stream>>>(mean, mean_cnt_elems);   // mean | cnt are contiguous
    k_build_w<<<(RR * FF * FF + 255) / 256, 256, 0, stream>>>(basis, comp, wst);
    k_count<<<(EE + 255) / 256, 256, 0, stream>>>(dstv, et, cnt);
    k_scatter<<<(unsigned)(((size_t)EE * 32 + 255) / 256), 256, 0, stream>>>(x_in, srcv, dstv, et, mean);
    k_gemm<<<NN / 16, 256, 0, stream>>>(mean, x_in, cnt, wst, root, bias, h_out);
}

static inline void run_bn(float* h, const float* gamma, const float* beta,
                          float* stats, hipStream_t stream) {
    k_zero<<<1, 256, 0, stream>>>(stats, 256);
    k_bn_stats<<<1024, 128, 0, stream>>>(h, stats);
    k_bn_apply<<<(unsigned)(((size_t)NN * FF + 255) / 256), 256, 0, stream>>>(h, stats, gamma, beta);
}

extern "C" void kernel_launch(void* const* d_in, const int* in_sizes, int n_in,
                              void* d_out, int out_size, void* d_ws, size_t ws_size,
                              hipStream_t stream) {
    (void)in_sizes; (void)n_in; (void)out_size; (void)ws_size;

    const float* x      = (const float*)d_in[0];
    const int*   eidx   = (const int*)d_in[1];          // [2,E]
    const int*   etype  = (const int*)d_in[2];
    const float* basis1 = (const float*)d_in[3];
    const float* comp1  = (const float*)d_in[4];
    const float* root1  = (const float*)d_in[5];
    const float* bias1  = (const float*)d_in[6];
    const float* basis2 = (const float*)d_in[7];
    const float* comp2  = (const float*)d_in[8];
    const float* root2  = (const float*)d_in[9];
    const float* bias2  = (const float*)d_in[10];
    const float* basis3 = (const float*)d_in[11];
    const float* comp3  = (const float*)d_in[12];
    const float* root3  = (const float*)d_in[13];
    const float* bias3  = (const float*)d_in[14];
    const float* bn1g   = (const float*)d_in[15];
    const float* bn1b   = (const float*)d_in[16];
    const float* bn2g   = (const float*)d_in[17];
    const float* bn2b   = (const float*)d_in[18];

    const int* srcv = eidx;          // edge_index[0]
    const int* dstv = eidx + EE;     // edge_index[1]

    // ---- workspace layout (floats) ----
    float* ws    = (float*)d_ws;
    float* mean  = ws;                                   // N*R*128 = 102,400,000
    float* cnt   = mean + (size_t)NN * RR * FF;          // N*R     =     800,000
    float* stats = cnt + (size_t)NN * RR;                // 256
    float* wst   = stats + 256;                          // 1024*128 =    131,072
    float* h1    = wst + (size_t)KA * FF;                // N*128   =  12,800,000
    float* h2    = h1 + (size_t)NN * FF;                 // N*128   =  12,800,000

    float* outf = (float*)d_out;

    // layer 1
    run_layer(x, basis1, comp1, root1, bias1, h1, srcv, dstv, etype, mean, cnt, wst, stream);
    run_bn(h1, bn1g, bn1b, stats, stream);
    // layer 2
    run_layer(h1, basis2, comp2, root2, bias2, h2, srcv, dstv, etype, mean, cnt, wst, stream);
    run_bn(h2, bn2g, bn2b, stats, stream);
    // layer 3 -> d_out, then row L2 normalize
    run_layer(h2, basis3, comp3, root3, bias3, outf, srcv, dstv, etype, mean, cnt, wst, stream);
    k_l2norm<<<NN, 128, 0, stream>>>(outf);
}